// ChildSumLSTMLayerWithEmbedding_13683765805738
// MI455X (gfx1250) — compile-verified
//
#include <hip/hip_runtime.h>

// ---------------------------------------------------------------------------
// Child-Sum Tree-LSTM, D=6 depths, N=32768 nodes, K=16 children, d=64.
//
// Algebraic rewrite: h_prev[idx] @ U_f == (h_prev @ U_f)[idx], so the per-child
// 64x64 GEMM over N*K rows becomes one GEMM over N+1 rows + a gather of its
// result (16x fewer FLOPs on the dominant term). All state tables (~8.4 MB)
// live in MI455X's 192 MB L2, so gathers are L2-bandwidth bound, not HBM.
// All dense math uses V_WMMA_F32_16X16X4_F32 (native CDNA5 fp32 matrix op,
// wave32, full fp32 precision matching the reference).
// ---------------------------------------------------------------------------

typedef __attribute__((ext_vector_type(2))) float v2f;
typedef __attribute__((ext_vector_type(8))) float v8f;

#define D_DEPTH 6
#define N_NODES 32768
#define K_CHILD 16
#define HDIM    64
#define NP      32784   // (N_NODES + 1) padded up to a multiple of 16 rows

__device__ __forceinline__ float fast_sigmoid(float x) {
    return 1.0f / (1.0f + __expf(-x));
}
__device__ __forceinline__ float fast_tanh(float x) {
    // tanh(x) = 1 - 2/(exp(2x)+1); saturates correctly at +-1 for |x| large
    return 1.0f - 2.0f / (__expf(2.0f * x) + 1.0f);
}

// A-matrix fragments for V_WMMA_F32_16X16X4_F32 over K=64 (16 k-steps).
// ISA layout (16x4 f32 A): lanes 0-15 hold {K=k0, K=k0+1} in the VGPR pair,
// lanes 16-31 hold {K=k0+2, K=k0+3}. rowp points at A[lane&15][0].
__device__ __forceinline__ void load_a16(const float* __restrict__ rowp,
                                         int lane, v2f a[16]) {
    const int co = 2 * (lane >> 4);
#pragma unroll
    for (int k = 0; k < 16; ++k)
        a[k] = *(const v2f*)(rowp + 4 * k + co);
}

// B-matrix fragment (4x16 f32 B, row-major source [64, ldb]):
// VGPR pair holds rows {k0, k0+1} for lanes 0-15 and {k0+2, k0+3} for 16-31,
// column = col + (lane & 15).
__device__ __forceinline__ v2f load_b(const float* __restrict__ B, int ldb,
                                      int kbase, int lane, int col) {
    const int kr = kbase + 2 * (lane >> 4);
    const int c  = col + (lane & 15);
    v2f b;
    b.x = B[kr * ldb + c];
    b.y = B[(kr + 1) * ldb + c];
    return b;
}

#define WMMA_F32(acc, a, b)                                                    \
    acc = __builtin_amdgcn_wmma_f32_16x16x4_f32(false, a, false, b, (short)0,  \
                                                acc, false, false)

// ---------------------------------------------------------------------------
// K1: Wx[n, 0:256] = E[labels[n]] @ W_w + W_b   (one wave = 16 rows x 64 cols)
// grid = (N/16, 4), block = 32
// ---------------------------------------------------------------------------
__global__ void embed_wx_kernel(const int* __restrict__ labels,
                                const float* __restrict__ E,
                                const float* __restrict__ Ww,   // [64,256]
                                const float* __restrict__ Wb,   // [256]
                                float* __restrict__ Wx) {       // [N,256]
    const int tile = blockIdx.x;
    const int cg   = blockIdx.y;          // column group: 64 cols each
    const int lane = threadIdx.x;
    const int row  = tile * 16 + (lane & 15);
    const int lab  = labels[row];

    v2f a[16];
    load_a16(E + (long long)lab * HDIM, lane, a);

    v8f acc[4] = {};
    const int colbase = cg * 64;
#pragma unroll
    for (int k = 0; k < 16; ++k) {
#pragma unroll
        for (int ct = 0; ct < 4; ++ct) {
            v2f b = load_b(Ww, 256, 4 * k, lane, colbase + ct * 16);
            WMMA_F32(acc[ct], a[k], b);
        }
    }
    // C/D layout: VGPR v -> M = 8*(lane>>4) + v, N = lane & 15
    const int rbase = tile * 16 + (lane >> 4) * 8;
    const int cb    = colbase + (lane & 15);
#pragma unroll
    for (int ct = 0; ct < 4; ++ct) {
#pragma unroll
        for (int v = 0; v < 8; ++v) {
            const int r = rbase + v;
            const int c = cb + ct * 16;
            Wx[r * 256 + c] = acc[ct][v] + Wb[c];
        }
    }
}

// ---------------------------------------------------------------------------
// K2: G[0:NP, :] = H[0:NP, :] @ U_f    (row 0 / padding rows of H are zero)
// grid = NP/16, block = 32
// ---------------------------------------------------------------------------
__global__ void uf_gemm_kernel(const float* __restrict__ H,   // [NP,64]
                               const float* __restrict__ Uf,  // [64,64]
                               float* __restrict__ G) {       // [NP,64]
    const int tile = blockIdx.x;
    const int lane = threadIdx.x;
    const int row  = tile * 16 + (lane & 15);

    v2f a[16];
    load_a16(H + row * HDIM, lane, a);

    v8f acc[4] = {};
#pragma unroll
    for (int k = 0; k < 16; ++k) {
#pragma unroll
        for (int ct = 0; ct < 4; ++ct) {
            v2f b = load_b(Uf, 64, 4 * k, lane, ct * 16);
            WMMA_F32(acc[ct], a[k], b);
        }
    }
    const int rbase = tile * 16 + (lane >> 4) * 8;
    const int cb    = lane & 15;
#pragma unroll
    for (int ct = 0; ct < 4; ++ct) {
#pragma unroll
        for (int v = 0; v < 8; ++v)
            G[(rbase + v) * HDIM + ct * 16 + cb] = acc[ct][v];
    }
}

// ---------------------------------------------------------------------------
// K3: per node: h_sum = sum_k h_prev[idx_k]; branch_f = sum_k sigmoid(W_f_x +
// G[idx_k]) * c_prev[idx_k]   (idx_k == 0 is masked). One thread = 4 dims of
// one node; a half-wave covers one full 64-float row -> coalesced L2 gathers.
// grid = N*16/256, block = 256
// ---------------------------------------------------------------------------
__global__ void child_reduce_kernel(const int* __restrict__ idx,    // [N,K]
                                    const float* __restrict__ hprev,// [NP,64]
                                    const float* __restrict__ cprev,// [NP,64]
                                    const float* __restrict__ G,    // [NP,64]
                                    const float* __restrict__ Wx,   // [N,256]
                                    float* __restrict__ hsum,       // [N,64]
                                    float* __restrict__ bf) {       // [N,64]
    const int t   = blockIdx.x * blockDim.x + threadIdx.x;
    const int n   = t >> 4;
    const int dim = (t & 15) << 2;

    const float4 wf = *(const float4*)(Wx + n * 256 + dim);  // f-gate x-part
    float4 hs = make_float4(0.f, 0.f, 0.f, 0.f);
    float4 ba = make_float4(0.f, 0.f, 0.f, 0.f);
    const int* ip = idx + n * K_CHILD;
#pragma unroll
    for (int k = 0; k < K_CHILD; ++k) {
        const int ci = ip[k];
        if (ci != 0) {
            const float4 h = *(const float4*)(hprev + ci * HDIM + dim);
            const float4 c = *(const float4*)(cprev + ci * HDIM + dim);
            const float4 g = *(const float4*)(G + ci * HDIM + dim);
            hs.x += h.x; hs.y += h.y; hs.z += h.z; hs.w += h.w;
            ba.x += fast_sigmoid(wf.x + g.x) * c.x;
            ba.y += fast_sigmoid(wf.y + g.y) * c.y;
            ba.z += fast_sigmoid(wf.z + g.z) * c.z;
            ba.w += fast_sigmoid(wf.w + g.w) * c.w;
        }
    }
    *(float4*)(hsum + n * HDIM + dim) = hs;
    *(float4*)(bf + n * HDIM + dim)   = ba;
}

// ---------------------------------------------------------------------------
// K4: iuo = h_sum @ U_iuo (64 -> 192), then gates. One wave owns 16 rows and
// all 192 output columns (12 v8f accumulators) so i/u/o for a given (row,dim)
// land in the same lane -> gate math stays in registers.
// grid = N/16, block = 32
// ---------------------------------------------------------------------------
__global__ void gates_kernel(const float* __restrict__ hsum,   // [N,64]
                             const float* __restrict__ Uiuo,   // [64,192]
                             const float* __restrict__ Wx,     // [N,256]
                             const float* __restrict__ bf,     // [N,64]
                             float* __restrict__ hnext,        // [NP,64]
                             float* __restrict__ cnext,        // [NP,64]
                             float* __restrict__ out) {        // [N,64] or null
    const int tile = blockIdx.x;
    const int lane = threadIdx.x;
    const int row  = tile * 16 + (lane & 15);

    v2f a[16];
    load_a16(hsum + row * HDIM, lane, a);

    v8f acc[12] = {};
#pragma unroll
    for (int k = 0; k < 16; ++k) {
#pragma unroll
        for (int ct = 0; ct < 12; ++ct) {
            v2f b = load_b(Uiuo, 192, 4 * k, lane, ct * 16);
            WMMA_F32(acc[ct], a[k], b);
        }
    }
    const int rbase = tile * 16 + (lane >> 4) * 8;
    const int cb    = lane & 15;
#pragma unroll
    for (int ct = 0; ct < 4; ++ct) {
#pragma unroll
        for (int v = 0; v < 8; ++v) {
            const int r = rbase + v;
            const int c = ct * 16 + cb;            // dim 0..63
            const float* wxr = Wx + r * 256;
            const float ig = fast_sigmoid(acc[ct][v]     + wxr[64 + c]);
            const float ug = fast_tanh   (acc[ct + 4][v] + wxr[128 + c]);
            const float og = fast_sigmoid(acc[ct + 8][v] + wxr[192 + c]);
            const float nc = ig * ug + bf[r * HDIM + c];
            const float nh = og * fast_tanh(nc);
            cnext[(r + 1) * HDIM + c] = nc;        // row 0 stays the zero row
            hnext[(r + 1) * HDIM + c] = nh;
            if (out) out[r * HDIM + c] = nh;
        }
    }
}

// ---------------------------------------------------------------------------
// Host driver (graph-capture safe: only kernel launches + one hipMemsetAsync)
// ---------------------------------------------------------------------------
extern "C" void kernel_launch(void* const* d_in, const int* in_sizes, int n_in,
                              void* d_out, int out_size, void* d_ws,
                              size_t ws_size, hipStream_t stream) {
    const int*   labels = (const int*)d_in[0];   // [D,N]
    const int*   child  = (const int*)d_in[1];   // [D,N,K]
    const float* E      = (const float*)d_in[2]; // [V,64]
    const float* Ww     = (const float*)d_in[3]; // [64,256]
    const float* Wb     = (const float*)d_in[4]; // [256]
    const float* Uf     = (const float*)d_in[5]; // [64,64]
    const float* Uiuo   = (const float*)d_in[6]; // [64,192]
    float*       out    = (float*)d_out;         // [N,64]

    // Workspace carve-up (~92 MB of fp32)
    float* p   = (float*)d_ws;
    float* h0  = p; p += (size_t)NP * HDIM;
    float* c0  = p; p += (size_t)NP * HDIM;
    float* h1  = p; p += (size_t)NP * HDIM;
    float* c1  = p; p += (size_t)NP * HDIM;
    float* G   = p; p += (size_t)NP * HDIM;
    float* Wx  = p; p += (size_t)N_NODES * 256;
    float* hsm = p; p += (size_t)N_NODES * HDIM;
    float* bfv = p; p += (size_t)N_NODES * HDIM;

    // h0,c0,h1,c1,G are contiguous: zero them (zero row + padding rows persist)
    hipMemsetAsync(h0, 0, (size_t)5 * NP * HDIM * sizeof(float), stream);

    float* hprev = h0; float* cprev = c0;
    float* hnext = h1; float* cnext = c1;

    for (int d = 0; d < D_DEPTH; ++d) {
        embed_wx_kernel<<<dim3(N_NODES / 16, 4), 32, 0, stream>>>(
            labels + (size_t)d * N_NODES, E, Ww, Wb, Wx);

        child_reduce_kernel<<<(N_NODES * 16) / 256, 256, 0, stream>>>(
            child + (size_t)d * N_NODES * K_CHILD, hprev, cprev, G, Wx, hsm, bfv);

        gates_kernel<<<N_NODES / 16, 32, 0, stream>>>(
            hsm, Uiuo, Wx, bfv, hnext, cnext,
            (d == D_DEPTH - 1) ? out : (float*)nullptr);

        if (d < D_DEPTH - 1)
            uf_gemm_kernel<<<NP / 16, 32, 0, stream>>>(hnext, Uf, G);

        float* t;
        t = hprev; hprev = hnext; hnext = t;
        t = cprev; cprev = cnext; cnext = t;
    }
}